// TropConv2D_8091718386441
// MI455X (gfx1250) — compile-verified
//
#include <hip/hip_runtime.h>
#include <hip/hip_bf16.h>

// Tropical 3x3 conv as semiring GEMM: r[p,f] = patch[p] + w[p,f], out = max_p r
// - min_p r + bias.  The broadcast ADD is exactly D = A*B + C with A[:,0]=patch,
// B[0,:]=1, C=w-tile -> offload to v_wmma_f32_16x16x4_f32 (XDL pipe, co-executes
// with VALU); VALU does only max3/min3.  w is pre-swizzled into the WMMA C-layout
// and staged per block via CDNA5 async global->LDS DMA.

#define HWDIM 32
#define HOUT  30
#define NF    64
#define KKC   288
#define NCHUNK 18              // 288 / 16
#define WSZ_FLOATS (NCHUNK * 4 * 32 * 8)   // 18432 floats = 72 KB
#define MAIN_THREADS 768       // 24 wave32 = 6 pixels x 4 f-tiles

typedef float v8f  __attribute__((ext_vector_type(8)));
typedef float v4ff __attribute__((ext_vector_type(4)));
typedef float v2f  __attribute__((ext_vector_type(2)));

#if __has_builtin(__builtin_amdgcn_wmma_f32_16x16x4_f32)
#define HAVE_WMMA 1
#else
#define HAVE_WMMA 0
#endif

#if __has_builtin(__builtin_amdgcn_global_load_async_to_lds_b128) && \
    __has_builtin(__builtin_amdgcn_s_wait_asynccnt)
#define HAVE_ASYNC_LDS 1
typedef int v4i __attribute__((__vector_size__(4 * sizeof(int))));
typedef __attribute__((address_space(1))) v4i glb_v4i;
typedef __attribute__((address_space(3))) v4i lds_v4i;
#else
#define HAVE_ASYNC_LDS 0
#endif

// ---- pre-kernel: swizzle w[288][64] into WMMA C-operand layout --------------
// tile (chunk, ftl): per lane two b128 halves:
//   ws[(chunk*4+ftl)*256 + (i>>2)*128 + lane*4 + (i&3)] = w[p][f]
//   p = chunk*16 + (lane<16 ? i : i+8),  f = ftl*16 + (lane&15),  i = 0..7
__global__ void swizzle_w_kernel(const float* __restrict__ w,
                                 float* __restrict__ ws) {
  int j = blockIdx.x * 256 + threadIdx.x;
  if (j >= WSZ_FLOATS) return;
  int q    = j & 3;
  int lane = (j >> 2) & 31;
  int h    = (j >> 7) & 1;
  int ftl  = (j >> 8) & 3;
  int chunk = j >> 10;
  int i = h * 4 + q;
  int p = chunk * 16 + ((lane < 16) ? i : i + 8);
  int f = ftl * 16 + (lane & 15);
  ws[j] = w[p * NF + f];
}

// x fetch for A-operand: A[m,0] = patch[chunk*16+m] on lanes 0-15, else 0
__device__ __forceinline__ float xfetch(const float* xb, int chunk, int lane) {
  const int tap = chunk >> 1, dr = tap / 3, dc = tap % 3;
  const int off = (dr * HWDIM + dc) * 32 + (chunk & 1) * 16 + (lane & 15);
  float v = xb[off];                       // lanes 16-31 read same addrs (coalesced)
  return (lane < 16) ? v : 0.0f;           // v_cndmask, no divergence
}

template <bool SWZ>
__device__ __forceinline__ v8f load_ctile(const float* lds_w, const float* w,
                                          int chunk, int ftl, int lane) {
  if (SWZ) {
    const float* tb = lds_w + (chunk * 4 + ftl) * 256 + lane * 4;
    v4ff lo = *(const v4ff*)(tb);          // ds_load_b128, conflict-free
    v4ff hi = *(const v4ff*)(tb + 128);
    return __builtin_shufflevector(lo, hi, 0, 1, 2, 3, 4, 5, 6, 7);
  } else {
    v8f c;
    const int col = ftl * 16 + (lane & 15);
    const int r0  = chunk * 16 + ((lane < 16) ? 0 : 8);
#pragma unroll
    for (int i = 0; i < 8; ++i) c[i] = w[(r0 + i) * NF + col];
    return c;
  }
}

template <bool SWZ>
__global__ __launch_bounds__(MAIN_THREADS)
void trop_wmma_kernel(const float* __restrict__ x,
                      const float* __restrict__ w,
                      const float* __restrict__ wsz,
                      const float* __restrict__ bias,
                      float* __restrict__ out) {
  __shared__ __align__(16) float lds_w[WSZ_FLOATS];  // 72 KB swizzled w

  const int tid  = threadIdx.x;
  const int lane = tid & 31;
  const int wave = tid >> 5;
  const int ftl  = wave & 3;                 // 16-filter tile 0..3
  const int pix  = wave >> 2;                // 0..5
  const int blk  = blockIdx.x;
  const int wg    = blk % 5;                 // wo group (6 pixels each)
  const int rowid = blk / 5;                 // b*30 + ho
  const int b  = rowid / HOUT;
  const int ho = rowid % HOUT;
  const int wo = wg * 6 + pix;

  if (SWZ) {
#if HAVE_ASYNC_LDS
    // async global->LDS DMA: 4608 b128 = exactly 6 per thread (EXEC stays full)
#pragma unroll
    for (int k = 0; k < 6; ++k) {
      int i = tid + k * MAIN_THREADS;
      __builtin_amdgcn_global_load_async_to_lds_b128(
          (glb_v4i*)(wsz + i * 4), (lds_v4i*)(lds_w + i * 4), 0, 0);
    }
    __builtin_amdgcn_s_wait_asynccnt(0);
#else
#pragma unroll
    for (int k = 0; k < 6; ++k) {
      int i = tid + k * MAIN_THREADS;
      ((float4*)lds_w)[i] = ((const float4*)wsz)[i];
    }
#endif
    __syncthreads();
  }

  const float* xb = x + (((size_t)b * HWDIM + ho) * HWDIM + wo) * 32;

  const v2f Bv = {(lane < 16) ? 1.0f : 0.0f, 0.0f};  // B[0,:]=1, rest 0
  v8f mx, mn;
#pragma unroll
  for (int i = 0; i < 8; ++i) {
    mx[i] = -__builtin_inff();
    mn[i] =  __builtin_inff();
  }

#pragma unroll
  for (int cc = 0; cc < NCHUNK; cc += 2) {
    v8f C1 = load_ctile<SWZ>(lds_w, w, cc + 0, ftl, lane);
    v8f C2 = load_ctile<SWZ>(lds_w, w, cc + 1, ftl, lane);
    float xa1 = xfetch(xb, cc + 0, lane);
    float xa2 = xfetch(xb, cc + 1, lane);
#if HAVE_WMMA
    v2f A1 = {xa1, 0.0f};
    v2f A2 = {xa2, 0.0f};
    // D[p,f] = patch[p]*1 + w[p,f]  (broadcast add on the matrix pipe)
    v8f D1 = __builtin_amdgcn_wmma_f32_16x16x4_f32(
        false, A1, false, Bv, (short)0, C1, false, false);
    v8f D2 = __builtin_amdgcn_wmma_f32_16x16x4_f32(
        false, A2, false, Bv, (short)0, C2, false, false);
#else
    v8f D1, D2;
#pragma unroll
    for (int i = 0; i < 8; ++i) {
      const int idx = (lane < 16) ? i : i + 8;
      const int t1 = cc >> 1, t2 = (cc + 1) >> 1;
      D1[i] = C1[i] + xb[((t1 / 3) * HWDIM + (t1 % 3)) * 32 + (cc & 1) * 16 + idx];
      D2[i] = C2[i] + xb[((t2 / 3) * HWDIM + (t2 % 3)) * 32 + ((cc + 1) & 1) * 16 + idx];
    }
#endif
#pragma unroll
    for (int i = 0; i < 8; ++i) {           // v_max3 / v_min3
      mx[i] = fmaxf(mx[i], fmaxf(D1[i], D2[i]));
      mn[i] = fminf(mn[i], fminf(D1[i], D2[i]));
    }
  }

  // per-lane tree over 8 regs (M=0..7 / 8..15 per half)
  float m = fmaxf(fmaxf(fmaxf(mx[0], mx[1]), fmaxf(mx[2], mx[3])),
                  fmaxf(fmaxf(mx[4], mx[5]), fmaxf(mx[6], mx[7])));
  float n = fminf(fminf(fminf(mn[0], mn[1]), fminf(mn[2], mn[3])),
                  fminf(fminf(mn[4], mn[5]), fminf(mn[6], mn[7])));
  // combine lane halves (M 0-7 vs 8-15)
  m = fmaxf(m, __shfl_xor(m, 16, 32));
  n = fminf(n, __shfl_xor(n, 16, 32));

  const int f = ftl * 16 + (lane & 15);
  const float res = m - n + bias[f];
  if (lane < 16)
    out[((size_t)rowid * HOUT + wo) * NF + f] = res;
}

extern "C" void kernel_launch(void* const* d_in, const int* in_sizes, int n_in,
                              void* d_out, int out_size, void* d_ws, size_t ws_size,
                              hipStream_t stream) {
  const float* x    = (const float*)d_in[0];  // [8,32,32,32]
  const float* w    = (const float*)d_in[1];  // [288,64]
  const float* bias = (const float*)d_in[2];  // [64]
  float* out = (float*)d_out;                 // [8,30,30,64]

  dim3 grid(8 * HOUT * 5);                    // 1200 blocks: 6-pixel groups
  dim3 block(MAIN_THREADS);                   // 24 wave32

  if (ws_size >= WSZ_FLOATS * sizeof(float)) {
    float* ws = (float*)d_ws;
    swizzle_w_kernel<<<(WSZ_FLOATS + 255) / 256, 256, 0, stream>>>(w, ws);
    trop_wmma_kernel<true><<<grid, block, 0, stream>>>(x, w, ws, bias, out);
  } else {
    trop_wmma_kernel<false><<<grid, block, 0, stream>>>(x, w, nullptr, bias, out);
  }
}